// SingleTaskGAT_9612136808655
// MI455X (gfx1250) — compile-verified
//
#include <hip/hip_runtime.h>

// ---------------------------------------------------------------------------
// GAT forward for MI455X (gfx1250, wave32).
// GEMMs use V_WMMA_F32_16X16X4_F32 (fp32 matrix pipe, full precision).
// Edge softmax / scatter use L2 atomics (whole working set is L2-resident).
// ---------------------------------------------------------------------------

#define NNODES  50000
#define NEDGES  800000
#define EHTOT   (NEDGES + NNODES)   // edges incl. self loops = 850000
#define HID     64
#define NHEADS  4
#define CHH     16
#define NLAYERS 4

typedef __attribute__((ext_vector_type(2))) float v2f;
typedef __attribute__((ext_vector_type(8))) float v8f;

// ---- ordered-float mapping for atomic max over signed floats ----
__device__ __forceinline__ unsigned f2ord(float f) {
  unsigned u = __float_as_uint(f);
  return (u & 0x80000000u) ? ~u : (u | 0x80000000u);
}
__device__ __forceinline__ float ord2f(unsigned u) {
  return (u & 0x80000000u) ? __uint_as_float(u & 0x7fffffffu)
                           : __uint_as_float(~u);
}

__global__ void zero_kernel(float* p, long n) {
  long t = (long)blockIdx.x * blockDim.x + threadIdx.x;
  if (t < n) p[t] = 0.0f;
}

// ---------------------------------------------------------------------------
// C[nrows,64] = A[nrows,K] @ B[K,64] (+ bias).  One wave per 16-row tile,
// 4 column tiles of 16 held in v8f accumulators, K stepped by 4 per WMMA.
// A fragment: row = lane&15, kb = (lane>>4)*2 -> {A[row][k+kb], A[row][k+kb+1]}
// B fragment: col = lane&15            -> {B[k+kb][col], B[k+kb+1][col]}
// D layout:   VGPR i, lanes0-15 -> row i, lanes16-31 -> row i+8, col = lane&15
// ---------------------------------------------------------------------------
template <int K>
__global__ void gemm_n64_wmma(const float* __restrict__ A,
                              const float* __restrict__ B,
                              const float* __restrict__ bias,
                              float* __restrict__ C, int nrows) {
  const int lane = threadIdx.x & 31;
  const int tile = blockIdx.x * (blockDim.x >> 5) + (threadIdx.x >> 5);
  if (tile * 16 >= nrows) return;            // wave-uniform: EXEC stays all-1s
  const int row0 = tile * 16;
  const int r    = lane & 15;
  const int kb   = (lane >> 4) << 1;
  const float* Arow = A + (long)(row0 + r) * K;

  v8f acc0 = {}, acc1 = {}, acc2 = {}, acc3 = {};
  for (int k0 = 0; k0 < K; k0 += 4) {
    v2f a;
    a.x = Arow[k0 + kb];
    a.y = Arow[k0 + kb + 1];
    const float* B0 = B + (long)(k0 + kb) * HID + r;
    const float* B1 = B0 + HID;
    v2f b0, b1, b2, b3;
    b0.x = B0[0];  b0.y = B1[0];
    b1.x = B0[16]; b1.y = B1[16];
    b2.x = B0[32]; b2.y = B1[32];
    b3.x = B0[48]; b3.y = B1[48];
    acc0 = __builtin_amdgcn_wmma_f32_16x16x4_f32(false, a, false, b0, (short)0, acc0, false, false);
    acc1 = __builtin_amdgcn_wmma_f32_16x16x4_f32(false, a, false, b1, (short)0, acc1, false, false);
    acc2 = __builtin_amdgcn_wmma_f32_16x16x4_f32(false, a, false, b2, (short)0, acc2, false, false);
    acc3 = __builtin_amdgcn_wmma_f32_16x16x4_f32(false, a, false, b3, (short)0, acc3, false, false);
  }

  const int mrow = row0 + ((lane >> 4) << 3);
  const float bv0 = bias ? bias[r]      : 0.0f;
  const float bv1 = bias ? bias[r + 16] : 0.0f;
  const float bv2 = bias ? bias[r + 32] : 0.0f;
  const float bv3 = bias ? bias[r + 48] : 0.0f;
#pragma unroll
  for (int i = 0; i < 8; ++i) {
    float* crow = C + (long)(mrow + i) * HID;
    crow[r]      = acc0[i] + bv0;
    crow[r + 16] = acc1[i] + bv1;
    crow[r + 32] = acc2[i] + bv2;
    crow[r + 48] = acc3[i] + bv3;
  }
}

// a_s[n,h] = sum_c xp[n,h*16+c]*att_src[h,c] ; likewise a_d
__global__ void att_coef_kernel(const float* __restrict__ xp,
                                const float* __restrict__ att_s,
                                const float* __restrict__ att_d,
                                float* __restrict__ a_s,
                                float* __restrict__ a_d) {
  int t = blockIdx.x * blockDim.x + threadIdx.x;
  if (t >= NNODES * NHEADS) return;
  int node = t >> 2, hh = t & 3;
  const float* v = xp + (long)node * HID + hh * CHH;
  float ss = 0.f, dd = 0.f;
#pragma unroll
  for (int c = 0; c < CHH; ++c) {
    float x = v[c];
    ss += x * att_s[hh * CHH + c];
    dd += x * att_d[hh * CHH + c];
  }
  a_s[t] = ss;
  a_d[t] = dd;
}

__device__ __forceinline__ void edge_sd(int e, const int* esrc, const int* edst,
                                        int& s, int& d) {
  if (e < NEDGES) { s = esrc[e]; d = edst[e]; }
  else            { s = d = e - NEDGES; }     // self loops
}

__global__ void edge_max_kernel(const int* __restrict__ esrc,
                                const int* __restrict__ edst,
                                const float* __restrict__ a_s,
                                const float* __restrict__ a_d,
                                unsigned* __restrict__ m_ord) {
  int t = blockIdx.x * blockDim.x + threadIdx.x;
  if (t >= EHTOT * NHEADS) return;
  int e = t >> 2, hh = t & 3, s, d;
  edge_sd(e, esrc, edst, s, d);
  float v = a_s[s * NHEADS + hh] + a_d[d * NHEADS + hh];
  v = v > 0.f ? v : 0.2f * v;                 // leaky_relu
  atomicMax(&m_ord[d * NHEADS + hh], f2ord(v));
}

__global__ void edge_sum_kernel(const int* __restrict__ esrc,
                                const int* __restrict__ edst,
                                const float* __restrict__ a_s,
                                const float* __restrict__ a_d,
                                const unsigned* __restrict__ m_ord,
                                float* __restrict__ ssum) {
  int t = blockIdx.x * blockDim.x + threadIdx.x;
  if (t >= EHTOT * NHEADS) return;
  int e = t >> 2, hh = t & 3, s, d;
  edge_sd(e, esrc, edst, s, d);
  float v = a_s[s * NHEADS + hh] + a_d[d * NHEADS + hh];
  v = v > 0.f ? v : 0.2f * v;
  float p = __expf(v - ord2f(m_ord[d * NHEADS + hh]));
  atomicAdd(&ssum[d * NHEADS + hh], p);
}

// one thread per (edge, head): 16 channels of scatter-add
__global__ void edge_agg_kernel(const int* __restrict__ esrc,
                                const int* __restrict__ edst,
                                const float* __restrict__ a_s,
                                const float* __restrict__ a_d,
                                const unsigned* __restrict__ m_ord,
                                const float* __restrict__ ssum,
                                const float* __restrict__ xp,
                                float* __restrict__ hagg) {
  int t = blockIdx.x * blockDim.x + threadIdx.x;
  if (t >= EHTOT * NHEADS) return;
  int e = t >> 2, hh = t & 3, s, d;
  edge_sd(e, esrc, edst, s, d);
  float v = a_s[s * NHEADS + hh] + a_d[d * NHEADS + hh];
  v = v > 0.f ? v : 0.2f * v;
  float alpha = __expf(v - ord2f(m_ord[d * NHEADS + hh])) /
                (ssum[d * NHEADS + hh] + 1e-16f);
  const float4* xv = (const float4*)(xp + (long)s * HID + hh * CHH);
  float* dstp = hagg + (long)d * HID + hh * CHH;
#pragma unroll
  for (int q = 0; q < 4; ++q) {
    float4 x4 = xv[q];
    atomicAdd(dstp + q * 4 + 0, x4.x * alpha);
    atomicAdd(dstp + q * 4 + 1, x4.y * alpha);
    atomicAdd(dstp + q * 4 + 2, x4.z * alpha);
    atomicAdd(dstp + q * 4 + 3, x4.w * alpha);
  }
}

// stats[0:64) = sum, [64:128) = sumsq
__global__ void bn_stats_kernel(const float* __restrict__ h,
                                float* __restrict__ stats) {
  int c = threadIdx.x & 63;
  float s = 0.f, q = 0.f;
  for (int r = blockIdx.x * 4 + (threadIdx.x >> 6); r < NNODES;
       r += gridDim.x * 4) {
    float x = h[(long)r * HID + c];
    s += x;
    q += x * x;
  }
  atomicAdd(&stats[c], s);
  atomicAdd(&stats[64 + c], q);
}

// stats[128:192) = mu, [192:256) = rstd
__global__ void bn_finalize_kernel(float* stats) {
  int c = threadIdx.x;
  if (c >= HID) return;
  float inv_n = 1.0f / (float)NNODES;
  float mu  = stats[c] * inv_n;
  float var = stats[64 + c] * inv_n - mu * mu;   // biased variance
  stats[128 + c] = mu;
  stats[192 + c] = rsqrtf(var + 1e-5f);
}

__global__ void bn_elu_kernel(const float* __restrict__ hagg,
                              const float* __restrict__ stats,
                              const float* __restrict__ gamma,
                              const float* __restrict__ beta,
                              float* __restrict__ hout) {
  int t = blockIdx.x * blockDim.x + threadIdx.x;
  if (t >= NNODES * HID) return;
  int c = t & 63;
  float x = (hagg[t] - stats[128 + c]) * stats[192 + c] * gamma[c] + beta[c];
  hout[t] = x > 0.f ? x : (__expf(x) - 1.0f);   // ELU
}

__global__ void pool_kernel(const float* __restrict__ h,
                            const int* __restrict__ batch,
                            float* __restrict__ g) {
  int t = blockIdx.x * blockDim.x + threadIdx.x;
  if (t >= NNODES * HID) return;
  int node = t >> 6, c = t & 63;
  atomicAdd(&g[(long)batch[node] * HID + c], h[t]);
}

// one wave32 per graph: hidden = relu(g@W1+b1); out = hidden@W2 + b2
__global__ void head_mlp_kernel(const float* __restrict__ g,
                                const float* __restrict__ W1,
                                const float* __restrict__ b1,
                                const float* __restrict__ W2,
                                const float* __restrict__ b2,
                                float* __restrict__ out) {
  int lane = threadIdx.x & 31;
  int gi = blockIdx.x * (blockDim.x >> 5) + (threadIdx.x >> 5);
  if (gi >= 256) return;
  const float* gr = g + (long)gi * HID;
  float partial = 0.f;
#pragma unroll
  for (int jj = 0; jj < 2; ++jj) {
    int j = lane * 2 + jj;
    float tacc = b1[j];
    for (int k = 0; k < HID; ++k) tacc += gr[k] * W1[k * HID + j];
    tacc = tacc > 0.f ? tacc : 0.f;
    partial += tacc * W2[j];
  }
  for (int off = 16; off; off >>= 1) partial += __shfl_xor(partial, off, 32);
  if (lane == 0) out[gi] = partial + b2[0];
}

// ---------------------------------------------------------------------------
static inline unsigned cdiv(long a, long b) { return (unsigned)((a + b - 1) / b); }

extern "C" void kernel_launch(void* const* d_in, const int* in_sizes, int n_in,
                              void* d_out, int out_size, void* d_ws, size_t ws_size,
                              hipStream_t stream) {
  const float* x        = (const float*)d_in[0];
  const float* embed_W  = (const float*)d_in[1];
  const float* embed_b  = (const float*)d_in[2];
  const float* gat_W    = (const float*)d_in[3];
  const float* att_src  = (const float*)d_in[4];
  const float* att_dst  = (const float*)d_in[5];
  // d_in[6] = gat_b: algebraically cancelled by the following BatchNorm.
  const float* bn_gamma = (const float*)d_in[7];
  const float* bn_beta  = (const float*)d_in[8];
  const float* fc1_W    = (const float*)d_in[9];
  const float* fc1_b    = (const float*)d_in[10];
  const float* fc2_W    = (const float*)d_in[11];
  const float* fc2_b    = (const float*)d_in[12];
  const int*   eidx     = (const int*)d_in[13];
  const int*   batch    = (const int*)d_in[14];
  float*       out      = (float*)d_out;

  const int* esrc = eidx;
  const int* edst = eidx + NEDGES;

  // ---- workspace layout (floats) ----
  float* ws    = (float*)d_ws;
  float* h     = ws;                         // N*64
  float* xp    = h    + (long)NNODES * HID;  // N*64
  float* hagg  = xp   + (long)NNODES * HID;  // N*64
  float* a_s   = hagg + (long)NNODES * HID;  // N*4
  float* a_d   = a_s  + (long)NNODES * NHEADS;
  unsigned* m_ord = (unsigned*)(a_d + (long)NNODES * NHEADS); // N*4
  float* ssum  = (float*)m_ord + (long)NNODES * NHEADS;       // N*4
  float* stats = ssum + (long)NNODES * NHEADS;                // 256
  float* gbuf  = stats + 256;                                 // 256*64

  const int ntiles = NNODES / 16;            // 3125 (exact)
  dim3 gemm_grid(cdiv(ntiles, 4)), gemm_blk(128);

  // embed: h = x @ embed_W + embed_b   (K = 128)
  gemm_n64_wmma<128><<<gemm_grid, gemm_blk, 0, stream>>>(x, embed_W, embed_b, h, NNODES);

  const long eh_threads = (long)EHTOT * NHEADS;   // 3.4M
  for (int l = 0; l < NLAYERS; ++l) {
    // xp = h @ gat_W[l]   (no bias)
    gemm_n64_wmma<64><<<gemm_grid, gemm_blk, 0, stream>>>(
        h, gat_W + (long)l * HID * HID, nullptr, xp, NNODES);

    att_coef_kernel<<<cdiv((long)NNODES * NHEADS, 256), 256, 0, stream>>>(
        xp, att_src + l * NHEADS * CHH, att_dst + l * NHEADS * CHH, a_s, a_d);

    // zero: hagg (N*64) and [m_ord | ssum | stats sum/sumsq] (contiguous)
    zero_kernel<<<cdiv((long)NNODES * HID, 256), 256, 0, stream>>>(hagg, (long)NNODES * HID);
    zero_kernel<<<cdiv((long)NNODES * NHEADS * 2 + 128, 256), 256, 0, stream>>>(
        (float*)m_ord, (long)NNODES * NHEADS * 2 + 128);

    edge_max_kernel<<<cdiv(eh_threads, 256), 256, 0, stream>>>(esrc, edst, a_s, a_d, m_ord);
    edge_sum_kernel<<<cdiv(eh_threads, 256), 256, 0, stream>>>(esrc, edst, a_s, a_d, m_ord, ssum);
    edge_agg_kernel<<<cdiv(eh_threads, 256), 256, 0, stream>>>(esrc, edst, a_s, a_d, m_ord, ssum, xp, hagg);

    bn_stats_kernel<<<256, 256, 0, stream>>>(hagg, stats);
    bn_finalize_kernel<<<1, 64, 0, stream>>>(stats);
    bn_elu_kernel<<<cdiv((long)NNODES * HID, 256), 256, 0, stream>>>(
        hagg, stats, bn_gamma + l * HID, bn_beta + l * HID, h);
  }

  // global add pool + MLP head
  zero_kernel<<<cdiv(256 * HID, 256), 256, 0, stream>>>(gbuf, 256 * HID);
  pool_kernel<<<cdiv((long)NNODES * HID, 256), 256, 0, stream>>>(h, batch, gbuf);
  head_mlp_kernel<<<128, 64, 0, stream>>>(gbuf, fc1_W, fc1_b, fc2_W, fc2_b, out);
}